// SparseAdagrad_65214783422592
// MI455X (gfx1250) — compile-verified
//
#include <hip/hip_runtime.h>
#include <math.h>

// SparseAdagrad on MI455X (gfx1250, wave32).
// Memory-bound: ~2.1 GB moved -> ~90us at 23.3 TB/s. No matrix contraction
// exists in this op, so WMMA is inapplicable; the CDNA5 paths that matter are
// b128 VMEM, f32 global atomics (resolved in L2), global_prefetch_b8, and the
// async DMA engine: ~88% of table rows are untouched verbatim copies, which we
// route through GLOBAL_LOAD_ASYNC_TO_LDS_B128 / GLOBAL_STORE_ASYNC_FROM_LDS_B128
// (ASYNCcnt-tracked, no VGPR data movement).

#define LR_CONST  0.01f
#define EPS_CONST 1e-10f

constexpr int D              = 128;  // embedding dim (reference constant)
constexpr int WAVE           = 32;   // gfx1250 is wave32-only
constexpr int ROWS_PER_BLOCK = 8;    // 256 threads = 8 waves, 1 row per wave

// ---- CDNA5 async Global<->LDS DMA (guarded: fall back to direct copies) ----
// Probe round 2 showed both builtins exist with signature
//   (v4i __device__* /*AS1*/, v4i __shared__* /*AS3*/, imm offset, imm cpol)
// where v4i = int __attribute__((vector_size(16))).
#if __has_builtin(__builtin_amdgcn_global_load_async_to_lds_b128) && \
    __has_builtin(__builtin_amdgcn_global_store_async_from_lds_b128)
#define HAS_ASYNC_LDS 1
typedef int v4i_b128 __attribute__((vector_size(16)));
typedef __attribute__((address_space(1))) v4i_b128* async_gptr_t;
typedef __attribute__((address_space(3))) v4i_b128* async_lptr_t;
// (void*) first strips const in generic AS, then addrspace+type cast.
#define ASYNC_LOAD_B128(gsrc, ldst)                                          \
    __builtin_amdgcn_global_load_async_to_lds_b128(                          \
        (async_gptr_t)(void*)(gsrc), (async_lptr_t)(void*)(ldst), 0, 0)
#define ASYNC_STORE_B128(gdst, lsrc)                                         \
    __builtin_amdgcn_global_store_async_from_lds_b128(                       \
        (async_gptr_t)(void*)(gdst), (async_lptr_t)(void*)(lsrc), 0, 0)
#else
#define HAS_ASYNC_LDS 0
#endif

#if __has_builtin(__builtin_amdgcn_s_wait_asynccnt)
#define WAIT_ASYNC_0() __builtin_amdgcn_s_wait_asynccnt(0)
#else
#define WAIT_ASYNC_0() asm volatile("s_wait_asynccnt 0x0" ::: "memory")
#endif

// ---------------------------------------------------------------- K1: cnt = 0
__global__ void sa_zero_cnt(unsigned int* __restrict__ cnt, int V) {
    int i = blockIdx.x * blockDim.x + threadIdx.x;
    if (i < V) cnt[i] = 0u;
}

// ------------------------------------------------------- K2: cnt[idx[i]] += 1
__global__ void sa_count(const int* __restrict__ idx,
                         unsigned int* __restrict__ cnt, int N) {
    int i = blockIdx.x * blockDim.x + threadIdx.x;
    if (i < N) atomicAdd(&cnt[idx[i]], 1u);
}

// -------------------------------------- K3: gsum[row,:] = 0 for touched rows
// gsum aliases the state_new half of d_out (always V*D floats available).
__global__ void sa_zero_touched(const unsigned int* __restrict__ cnt,
                                float4* __restrict__ gsum4, int V) {
    int wave = threadIdx.x / WAVE;
    int lane = threadIdx.x % WAVE;
    int row  = blockIdx.x * ROWS_PER_BLOCK + wave;
    if (row >= V) return;
    if (cnt[row] != 0u) {                      // wave-uniform branch
        gsum4[(size_t)row * (D / 4) + lane] = make_float4(0.f, 0.f, 0.f, 0.f);
    }
}

// --------------------------------- K4: gsum[idx[i],:] += grad[i,:] (atomics)
// One wave per input row; each lane owns 4 consecutive floats.
__global__ void sa_scatter(const int* __restrict__ idx,
                           const float4* __restrict__ grad4,
                           float* __restrict__ gsum, int N) {
    int wave = threadIdx.x / WAVE;
    int lane = threadIdx.x % WAVE;
    int i    = blockIdx.x * ROWS_PER_BLOCK + wave;
    if (i >= N) return;
    int v = idx[i];                                     // wave-uniform
    float4 g = grad4[(size_t)i * (D / 4) + lane];       // global_load_b128
    float* dst = gsum + (size_t)v * D + (size_t)lane * 4;
    atomicAdd(dst + 0, g.x);                            // global_atomic_add_f32
    atomicAdd(dst + 1, g.y);
    atomicAdd(dst + 2, g.z);
    atomicAdd(dst + 3, g.w);
}

// ---------------------------------------------------------------- K5: update
// Touched (~12%): gm = gsum/cnt; state_new = state + gm^2;
//                 emb_new = emb - lr*gm/(sqrt(state_new)+eps)   (VGPR math)
// Untouched (~88%): bit-identical copy via async DMA bounce through LDS
//                 (global_load_async_to_lds_b128 -> s_wait_asynccnt ->
//                  global_store_async_from_lds_b128). Each lane uses only the
//                 LDS slot its own wave's async op wrote -> no barrier needed;
//                 S_ENDPGM's implicit wait-idle drains the trailing stores.
// out_state4 holds gsum on entry; each wave reads its own row then overwrites
// it with state_new (no cross-thread aliasing).
__global__ void sa_update(const unsigned int* __restrict__ cnt,
                          const float4* __restrict__ emb4,
                          const float4* __restrict__ state4,
                          float4* __restrict__ out_emb4,
                          float4* __restrict__ out_state4,
                          int V) {
#if HAS_ASYNC_LDS
    __shared__ float4 lds_e[ROWS_PER_BLOCK][WAVE];   // 4 KB
    __shared__ float4 lds_s[ROWS_PER_BLOCK][WAVE];   // 4 KB
#endif
    int wave = threadIdx.x / WAVE;
    int lane = threadIdx.x % WAVE;
    int row  = blockIdx.x * ROWS_PER_BLOCK + wave;
    if (row >= V) return;

    size_t base = (size_t)row * (D / 4) + lane;
    unsigned int c = cnt[row];                 // wave-uniform

    if (c == 0u) {
        // ---------- untouched row: verbatim copy ----------
#if HAS_ASYNC_LDS
        ASYNC_LOAD_B128(emb4 + base,   &lds_e[wave][lane]);
        ASYNC_LOAD_B128(state4 + base, &lds_s[wave][lane]);
        WAIT_ASYNC_0();   // async LDS writes may complete out of order vs reads
        ASYNC_STORE_B128(out_emb4 + base,   &lds_e[wave][lane]);
        ASYNC_STORE_B128(out_state4 + base, &lds_s[wave][lane]);
#else
        out_emb4[base]   = emb4[base];
        out_state4[base] = state4[base];
#endif
        return;
    }

    // ---------- touched row: Adagrad step in VGPRs ----------
    // Speculative prefetch one block of rows ahead (global_prefetch_b8).
    constexpr size_t AHEAD = (size_t)ROWS_PER_BLOCK * (D / 4);
    __builtin_prefetch((const char*)(emb4 + base + AHEAD), 0, 1);
    __builtin_prefetch((const char*)(state4 + base + AHEAD), 0, 1);

    float4 e = emb4[base];                     // global_load_b128
    float4 s = state4[base];                   // global_load_b128
    float4 g = out_state4[base];               // gsum row
    float inv_c = 1.0f / (float)c;

    float4 sn, en;
    {
        float gm;
        gm = g.x * inv_c; sn.x = s.x + gm * gm;
        en.x = e.x - LR_CONST * gm / (sqrtf(sn.x) + EPS_CONST);
        gm = g.y * inv_c; sn.y = s.y + gm * gm;
        en.y = e.y - LR_CONST * gm / (sqrtf(sn.y) + EPS_CONST);
        gm = g.z * inv_c; sn.z = s.z + gm * gm;
        en.z = e.z - LR_CONST * gm / (sqrtf(sn.z) + EPS_CONST);
        gm = g.w * inv_c; sn.w = s.w + gm * gm;
        en.w = e.w - LR_CONST * gm / (sqrtf(sn.w) + EPS_CONST);
    }

    out_emb4[base]   = en;                     // global_store_b128
    out_state4[base] = sn;                     // global_store_b128
}

// ---------------------------------------------------------------------------
extern "C" void kernel_launch(void* const* d_in, const int* in_sizes, int n_in,
                              void* d_out, int out_size, void* d_ws, size_t ws_size,
                              hipStream_t stream) {
    const int*   idx   = (const int*)  d_in[0];   // [N]
    const float* grad  = (const float*)d_in[1];   // [N, D]
    const float* emb   = (const float*)d_in[2];   // [V, D]
    const float* state = (const float*)d_in[3];   // [V, D]

    const int N = in_sizes[0];
    const int V = in_sizes[2] / D;

    float* out_emb   = (float*)d_out;                      // [V, D]
    float* out_state = out_emb + (size_t)V * D;            // [V, D] (gsum first)

    unsigned int* cnt = (unsigned int*)d_ws;               // V u32 (4 MB)

    const int row_blocks_V = (V + ROWS_PER_BLOCK - 1) / ROWS_PER_BLOCK;
    const int row_blocks_N = (N + ROWS_PER_BLOCK - 1) / ROWS_PER_BLOCK;

    sa_zero_cnt<<<(V + 255) / 256, 256, 0, stream>>>(cnt, V);
    sa_count<<<(N + 255) / 256, 256, 0, stream>>>(idx, cnt, N);
    sa_zero_touched<<<row_blocks_V, 256, 0, stream>>>(cnt, (float4*)out_state, V);
    sa_scatter<<<row_blocks_N, 256, 0, stream>>>(idx, (const float4*)grad,
                                                 out_state, N);
    sa_update<<<row_blocks_V, 256, 0, stream>>>(cnt,
                                                (const float4*)emb,
                                                (const float4*)state,
                                                (float4*)out_emb,
                                                (float4*)out_state, V);
}